// WaveletFeatureExtractor_39865886441541
// MI455X (gfx1250) — compile-verified
//
#include <hip/hip_runtime.h>
#include <hip/hip_bf16.h>

typedef float v2f __attribute__((ext_vector_type(2)));
typedef float v4f __attribute__((ext_vector_type(4)));
typedef float v8f __attribute__((ext_vector_type(8)));

// db4 filters, pre-reversed for jnp.convolve semantics: conv[k] = sum_j ext[k+j]*f[7-j]
__constant__ float LO_R[8] = {
     0.23037781330885523f,  0.7148465705525415f,   0.6308807679295904f,
    -0.02798376941698385f, -0.18703481171888114f,  0.030841381835986965f,
     0.032883011666982945f, -0.010597401784997278f };
__constant__ float HI_R[8] = {
    -0.010597401784997278f, -0.032883011666982945f, 0.030841381835986965f,
     0.18703481171888114f,  -0.02798376941698385f, -0.6308807679295904f,
     0.7148465705525415f,   -0.23037781330885523f };

// ---------------------------------------------------------------------------
// Kernel 1: 4-level db4 wavedec stats -> feats (128 x 10).
// One block (256 thr = 8 waves) per row. Signal ping-pongs in LDS.
// ---------------------------------------------------------------------------
__global__ __launch_bounds__(256) void wavelet_feats_k(
    const float* __restrict__ x, float* __restrict__ feats)
{
    __shared__ float buf0[4096];
    __shared__ float buf1[2051];
    __shared__ float acc[6];

    const int tid = threadIdx.x;
    const int row = blockIdx.x;
    const float* xr = x + (size_t)row * 4096;

    for (int i = tid; i < 4096; i += 256) buf0[i] = xr[i];
    __syncthreads();

    float* cur = buf0;
    float* nxt = buf1;
    int n = 4096;

    for (int lvl = 0; lvl < 4; ++lvl) {
        if (tid < 6) acc[tid] = 0.0f;
        __syncthreads();

        const int outLen = (n + 7) >> 1;   // pywt: floor((n + FLEN - 1)/2)
        float dAbs = 0.f, dSum = 0.f, dSq = 0.f;
        float aAbs = 0.f, aSum = 0.f, aSq = 0.f;

        for (int i = tid; i < outLen; i += 256) {
            const int k = 2 * i + 1;       // odd samples of full conv
            float lo = 0.f, hi = 0.f;
#pragma unroll
            for (int j = 0; j < 8; ++j) {
                const int t = k + j;       // index into sym-extended signal
                int src;
                if (t < 7)          src = 6 - t;           // left half-point reflect
                else if (t < n + 7) src = t - 7;           // body
                else                src = 2 * n + 6 - t;   // right reflect
                const float v = cur[src];
                lo = fmaf(v, LO_R[j], lo);
                hi = fmaf(v, HI_R[j], hi);
            }
            nxt[i] = lo;
            dAbs += fabsf(hi); dSum += hi; dSq += hi * hi;
            if (lvl == 3) { aAbs += fabsf(lo); aSum += lo; aSq += lo * lo; }
        }

        // wave32 reduce, then shared ds_add_f32 across the 8 waves
#pragma unroll
        for (int o = 16; o > 0; o >>= 1) {
            dAbs += __shfl_down(dAbs, o, 32);
            dSum += __shfl_down(dSum, o, 32);
            dSq  += __shfl_down(dSq,  o, 32);
            aAbs += __shfl_down(aAbs, o, 32);
            aSum += __shfl_down(aSum, o, 32);
            aSq  += __shfl_down(aSq,  o, 32);
        }
        if ((tid & 31) == 0) {
            atomicAdd(&acc[0], dAbs); atomicAdd(&acc[1], dSum); atomicAdd(&acc[2], dSq);
            if (lvl == 3) {
                atomicAdd(&acc[3], aAbs); atomicAdd(&acc[4], aSum); atomicAdd(&acc[5], aSq);
            }
        }
        __syncthreads();

        if (tid == 0) {
            const float len = (float)outLen;
            const float md  = acc[1] / len;
            const float vd  = fmaxf(acc[2] / len - md * md, 0.f);
            const int base  = 8 - 2 * lvl;       // cD1->8, cD2->6, cD3->4, cD4->2
            feats[row * 10 + base]     = acc[0] / len;
            feats[row * 10 + base + 1] = sqrtf(vd);
            if (lvl == 3) {                      // cA4 -> feats 0,1
                const float ma = acc[4] / len;
                const float va = fmaxf(acc[5] / len - ma * ma, 0.f);
                feats[row * 10 + 0] = acc[3] / len;
                feats[row * 10 + 1] = sqrtf(va);
            }
        }
        __syncthreads();                          // protects acc + buffer swap

        float* t = cur; cur = nxt; nxt = t;
        n = outLen;                               // 4096->2051->1029->518->262
    }
}

// ---------------------------------------------------------------------------
// Kernel 2: h = relu(feats @ W1 + b1), (128 x 512), K = 10. Trivial.
// ---------------------------------------------------------------------------
__global__ __launch_bounds__(512) void mlp1_k(
    const float* __restrict__ feats, const float* __restrict__ W1,
    const float* __restrict__ b1, float* __restrict__ h)
{
    __shared__ float f[10];
    const int row = blockIdx.x;
    const int j   = threadIdx.x;
    if (j < 10) f[j] = feats[row * 10 + j];
    __syncthreads();
    float s = b1[j];
#pragma unroll
    for (int k = 0; k < 10; ++k) s = fmaf(f[k], W1[k * 512 + j], s);
    h[row * 512 + j] = fmaxf(s, 0.f);
}

// ---------------------------------------------------------------------------
// Kernel 3: out_small = h @ W2 + b2 via V_WMMA_F32_16X16X4_F32.
// 256 tiles (8 M-tiles x 32 N-tiles), one wave per 16x16 tile, 128 K-steps.
// A layout: lane holds A[M = lane&15][K = 2*(lane>>4) + r], r = VGPR index.
// B layout: lane holds B[K = 2*(lane>>4) + r][N = lane&15].
// C/D layout: VGPR r holds M = r + 8*(lane>>4), N = lane&15.
// ---------------------------------------------------------------------------
__global__ __launch_bounds__(256) void mlp2_wmma_k(
    const float* __restrict__ h, const float* __restrict__ W2,
    const float* __restrict__ b2, float* __restrict__ out)
{
    const int lane = threadIdx.x & 31;
    const int wave = threadIdx.x >> 5;
    const int tile = blockIdx.x * 8 + wave;   // 0..255
    const int tM   = tile >> 5;               // 0..7
    const int tN   = tile & 31;               // 0..31
    const int m    = tM * 16 + (lane & 15);
    const int nn   = tN * 16 + (lane & 15);
    const int kb   = (lane >> 4) * 2;

    v8f c = {};
    for (int k0 = 0; k0 < 512; k0 += 4) {
        v2f a, b;
        a[0] = h[m * 512 + (k0 + kb)];
        a[1] = h[m * 512 + (k0 + kb + 1)];
        b[0] = W2[(k0 + kb)     * 512 + nn];
        b[1] = W2[(k0 + kb + 1) * 512 + nn];
        c = __builtin_amdgcn_wmma_f32_16x16x4_f32(
                /*neg_a=*/false, a, /*neg_b=*/false, b,
                /*c_mod=*/(short)0, c, /*reuse_a=*/false, /*reuse_b=*/false);
    }

    const float bias = b2[nn];
#pragma unroll
    for (int r = 0; r < 8; ++r) {
        const int mrow = tM * 16 + r + 8 * (lane >> 4);
        out[mrow * 512 + nn] = c[r] + bias;
    }
}

// ---------------------------------------------------------------------------
// Kernel 4: broadcast (128*512) -> (128, 512, 4096). 1 GiB of stores; this is
// the roofline kernel (~46 us at 23.3 TB/s). Coalesced 128-bit NT stores.
// ---------------------------------------------------------------------------
__global__ __launch_bounds__(256) void broadcast_k(
    const float* __restrict__ src, float* __restrict__ dst)
{
    const size_t row = blockIdx.x;            // 0..65535 == (b*512 + d)
    const float v = src[row];
    v4f vv = { v, v, v, v };
    v4f* base = (v4f*)(dst + row * 4096);
#pragma unroll
    for (int j = 0; j < 4; ++j)
        __builtin_nontemporal_store(vv, base + j * 256 + threadIdx.x);
}

// ---------------------------------------------------------------------------
extern "C" void kernel_launch(void* const* d_in, const int* in_sizes, int n_in,
                              void* d_out, int out_size, void* d_ws, size_t ws_size,
                              hipStream_t stream)
{
    const float* x  = (const float*)d_in[0];   // (128, 4096)
    const float* W1 = (const float*)d_in[1];   // (10, 512)
    const float* b1 = (const float*)d_in[2];   // (512,)
    const float* W2 = (const float*)d_in[3];   // (512, 512)
    const float* b2 = (const float*)d_in[4];   // (512,)
    float* out = (float*)d_out;                // (128, 512, 4096)

    float* ws    = (float*)d_ws;
    float* feats = ws;                         // 128*10   = 1280
    float* h     = ws + 1280;                  // 128*512  = 65536
    float* osml  = ws + 1280 + 65536;          // 128*512  = 65536

    wavelet_feats_k<<<128,   256, 0, stream>>>(x, feats);
    mlp1_k        <<<128,   512, 0, stream>>>(feats, W1, b1, h);
    mlp2_wmma_k   <<<32,    256, 0, stream>>>(h, W2, b2, osml);
    broadcast_k   <<<65536, 256, 0, stream>>>(osml, out);
}